// SNAP_GNN_85306640433747
// MI455X (gfx1250) — compile-verified
//
#include <hip/hip_runtime.h>

// ---------------------------------------------------------------------------
// MI455X (gfx1250) GCN pipeline.
//  - Dense GEMMs: V_WMMA_F32_16X16X4_F32 (fp32 matrix pipe, wave32), shapes
//    templated compile-time; K-loops fully unrolled; B operand K-pair
//    interleaved in LDS so each load is a single ds_load_b64 straight into
//    the WMMA source register pair; FULL-tile fast path (N%16==0) removes
//    all row masking from the hot loop.
//  - Graph aggregation: GLOBAL_ATOMIC_ADD_F32 scatter + global_prefetch on
//    the gather stream (the true bottleneck: ~1.5 GB irregular @ 23.3 TB/s).
// ---------------------------------------------------------------------------

typedef __attribute__((ext_vector_type(2))) float v2f;
typedef __attribute__((ext_vector_type(8))) float v8f;

#define TPB 256

static inline int cdiv_i(long a, long b) { return (int)((a + b - 1) / b); }

// ---------------- degree / normalization ----------------

__global__ void deg_init_kernel(float* __restrict__ a, float* __restrict__ b, int n) {
  int i = blockIdx.x * blockDim.x + threadIdx.x;
  if (i < n) { a[i] = 1.0f; b[i] = 1.0f; }   // 1.0 = self-loop
}

__global__ void deg_count_kernel(const int* __restrict__ dst, int e, float* __restrict__ deg) {
  int stride = gridDim.x * blockDim.x;
  for (int i = blockIdx.x * blockDim.x + threadIdx.x; i < e; i += stride)
    atomicAdd(&deg[dst[i]], 1.0f);
}

__global__ void deg_rsqrt_kernel(float* __restrict__ d, int n) {
  int i = blockIdx.x * blockDim.x + threadIdx.x;
  if (i < n) d[i] = rsqrtf(d[i]);            // deg >= 1 always (self-loop)
}

// ---------------- GCN aggregation ----------------

// out[v] = xw[v] * dinv[v]^2   (self-loop contribution, also initializes out)
__global__ void selfloop_kernel(const float* __restrict__ xw, const float* __restrict__ dinv,
                                float* __restrict__ out, int n, int d) {
  int i = blockIdx.x * blockDim.x + threadIdx.x;
  if (i < n * d) {
    int v = i / d;
    float w = dinv[v];
    out[i] = xw[i] * w * w;
  }
}

// out[dst] += dinv[src]*dinv[dst] * xw[src]  for every edge (atomic scatter-add)
__global__ void edge_scatter_kernel(const float* __restrict__ xw, const int* __restrict__ src,
                                    const int* __restrict__ dst, const float* __restrict__ dinv,
                                    float* __restrict__ out, int e, int d) {
  int stride = gridDim.x * blockDim.x;
  for (int i = blockIdx.x * blockDim.x + threadIdx.x; i < e; i += stride) {
    int s = src[i];
    int t = dst[i];
    int i2 = i + stride;
    if (i2 < e)  // hide gather latency of the next edge's source row
      __builtin_prefetch(&xw[(size_t)src[i2] * d], 0, 1);
    float nrm = dinv[s] * dinv[t];
    const float* xs = xw + (size_t)s * d;
    float* od = out + (size_t)t * d;
    for (int j = 0; j < d; ++j)
      atomicAdd(od + j, nrm * xs[j]);
  }
}

__global__ void bias_act_kernel(float* __restrict__ out, const float* __restrict__ b,
                                int n, int d, int relu) {
  int i = blockIdx.x * blockDim.x + threadIdx.x;
  if (i < n * d) {
    float v = out[i] + b[i % d];
    if (relu) v = fmaxf(v, 0.0f);
    out[i] = v;
  }
}

__global__ void concat_relu_kernel(const float* __restrict__ a, int da,
                                   const float* __restrict__ b, int db,
                                   float* __restrict__ out, int n) {
  int dc = da + db;
  int i = blockIdx.x * blockDim.x + threadIdx.x;
  if (i < n * dc) {
    int v = i / dc, c = i - v * dc;
    float x = (c < da) ? a[(size_t)v * da + c] : b[(size_t)v * db + (c - da)];
    out[i] = fmaxf(x, 0.0f);
  }
}

// ---------------- WMMA fp32 GEMM:  Y = act(X[n,DIN] @ W[DIN,DOUT] + b) ----------------
// One wave32 per 16-row node tile; NTILE 16-col accumulator tiles.
// A 16x4 f32 tile: lanes 0-15 hold K=k,k+1; lanes 16-31 hold K=k+2,k+3 (ISA 7.12.2).
// C/D: v8f, element i <-> row (i + 8*half), col = lane&15.
// W is stored K-pair interleaved in LDS: sW[(k/2)*2*CPAD + 2*c + {0,1}] so the
// B operand is one aligned contiguous v2f -> single ds_load_b64 into the WMMA
// source pair. FULL=true (n%16==0) removes all row masks from the hot loop.

template <int DIN, int DOUT, bool FULL>
__global__ __launch_bounds__(256) void wmma_gemm_kernel(
    const float* __restrict__ X, const float* __restrict__ W, const float* __restrict__ Bv,
    float* __restrict__ Y, int n, int relu) {
  constexpr int KPAD  = (DIN + 3) & ~3;
  constexpr int NTILE = (DOUT + 15) / 16;
  constexpr int CPAD  = NTILE * 16;

  __shared__ float sW[KPAD * CPAD];   // K-pair interleaved
  __shared__ float sB[CPAD];

  for (int idx = threadIdx.x; idx < (KPAD / 2) * CPAD; idx += TPB) {
    int p = idx / CPAD, c = idx - p * CPAD;
    int k0 = 2 * p, k1 = 2 * p + 1;
    sW[p * 2 * CPAD + 2 * c + 0] = (k0 < DIN && c < DOUT) ? W[k0 * DOUT + c] : 0.0f;
    sW[p * 2 * CPAD + 2 * c + 1] = (k1 < DIN && c < DOUT) ? W[k1 * DOUT + c] : 0.0f;
  }
  if (threadIdx.x < CPAD)
    sB[threadIdx.x] = (Bv != nullptr && threadIdx.x < DOUT) ? Bv[threadIdx.x] : 0.0f;
  __syncthreads();

  const int lane = threadIdx.x & 31;       // wave32
  const int hi   = lane >> 4;              // half-wave select
  const int m    = lane & 15;
  const int row0 = (blockIdx.x * 8 + (threadIdx.x >> 5)) * 16;
  if (row0 >= n) return;                   // wave-uniform exit

  int r = row0 + m;
  float rmask = 1.0f;
  if (!FULL) {                             // branchless row handling (partial tiles)
    rmask = (r < n) ? 1.0f : 0.0f;
    if (r >= n) r = n - 1;
  }
  const float* xrow = X + (size_t)r * DIN;

  v8f acc[NTILE] = {};

#pragma unroll
  for (int k = 0; k < KPAD; k += 4) {
    const int ka = k + 2 * hi;             // even; lanes 0-15: k,k+1; lanes 16-31: k+2,k+3
    v2f a;
    if (k + 3 < DIN) {                     // compile-time: full K-group
      if (FULL) { a.x = xrow[ka];         a.y = xrow[ka + 1]; }
      else      { a.x = xrow[ka] * rmask; a.y = xrow[ka + 1] * rmask; }
    } else {                               // tail K-group: clamp + select (no branches)
      int k0 = (ka < DIN) ? ka : 0;
      int k1 = (ka + 1 < DIN) ? ka + 1 : 0;
      a.x = ((ka     < DIN) ? rmask : 0.0f) * xrow[k0];
      a.y = ((ka + 1 < DIN) ? rmask : 0.0f) * xrow[k1];
    }
    const int pbase = (ka >> 1) * 2 * CPAD;   // K-pair row base in interleaved LDS
#pragma unroll
    for (int t = 0; t < NTILE; ++t) {
      v2f b = *(const v2f*)&sW[pbase + 2 * (t * 16 + m)];   // one ds_load_b64
      acc[t] = __builtin_amdgcn_wmma_f32_16x16x4_f32(
          /*neg_a=*/false, a, /*neg_b=*/false, b,
          /*c_mod=*/(short)0, acc[t], /*reuse_a=*/false, /*reuse_b=*/false);
    }
  }

#pragma unroll
  for (int t = 0; t < NTILE; ++t) {
#pragma unroll
    for (int i = 0; i < 8; ++i) {
      int rr = row0 + i + 8 * hi;
      int c  = t * 16 + m;
      bool cok = (t * 16 + 16 <= DOUT) ? true : (c < DOUT);   // folds for full tiles
      if ((FULL || rr < n) && cok) {
        float v = acc[t][i] + sB[c];
        if (relu) v = fmaxf(v, 0.0f);
        Y[(size_t)rr * DOUT + c] = v;
      }
    }
  }
}

template <int DIN, int DOUT>
static void run_gemm(const float* X, const float* W, const float* B, float* Y,
                     int n, int relu, hipStream_t s) {
  dim3 g(cdiv_i(cdiv_i(n, 16), 8)), blk(TPB);
  if ((n & 15) == 0)
    wmma_gemm_kernel<DIN, DOUT, true ><<<g, blk, 0, s>>>(X, W, B, Y, n, relu);
  else
    wmma_gemm_kernel<DIN, DOUT, false><<<g, blk, 0, s>>>(X, W, B, Y, n, relu);
}

// ---------------- driver ----------------

extern "C" void kernel_launch(void* const* d_in, const int* in_sizes, int n_in,
                              void* d_out, int out_size, void* d_ws, size_t ws_size,
                              hipStream_t stream) {
  (void)n_in; (void)out_size; (void)ws_size;

  const float* feat  = (const float*)d_in[0];
  const float* spat  = (const float*)d_in[1];
  const int*   fei   = (const int*)d_in[2];
  const int*   sei   = (const int*)d_in[3];
  const float* W_fc  = (const float*)d_in[4];  const float* b_fc  = (const float*)d_in[5];
  const float* W_cnn = (const float*)d_in[6];  const float* b_cnn = (const float*)d_in[7];
  const float* Wf1   = (const float*)d_in[8];  const float* bf1   = (const float*)d_in[9];
  const float* Wf2   = (const float*)d_in[10]; const float* bf2   = (const float*)d_in[11];
  const float* Ws1   = (const float*)d_in[12]; const float* bs1   = (const float*)d_in[13];
  const float* Ws2   = (const float*)d_in[14]; const float* bs2   = (const float*)d_in[15];
  const float* Wp1   = (const float*)d_in[16]; const float* bp1   = (const float*)d_in[17];
  const float* Wp2   = (const float*)d_in[18]; const float* bp2   = (const float*)d_in[19];

  const int N = in_sizes[0] / 31;
  const int E = in_sizes[2] / 2;
  const int* f_src = fei;     const int* f_dst = fei + E;
  const int* s_src = sei;     const int* s_dst = sei + E;

  // Workspace layout (floats): ~N*144 -> ~58 MB
  float* ws     = (float*)d_ws;
  float* dinv_f = ws;                 ws += N;
  float* dinv_s = ws;                 ws += N;
  float* bufA   = ws;                 ws += (size_t)N * 33;  // xw scratch
  float* bufF   = ws;                 ws += (size_t)N * 33;  // feat branch activation
  float* bufS   = ws;                 ws += (size_t)N * 32;  // spat branch activation
  float* bufC   = ws;                 ws += (size_t)N * 44;  // concat

  // --- symmetric degree normalization (both graphs) ---
  deg_init_kernel<<<cdiv_i(N, TPB), TPB, 0, stream>>>(dinv_f, dinv_s, N);
  deg_count_kernel<<<2048, TPB, 0, stream>>>(f_dst, E, dinv_f);
  deg_count_kernel<<<2048, TPB, 0, stream>>>(s_dst, E, dinv_s);
  deg_rsqrt_kernel<<<cdiv_i(N, TPB), TPB, 0, stream>>>(dinv_f, N);
  deg_rsqrt_kernel<<<cdiv_i(N, TPB), TPB, 0, stream>>>(dinv_s, N);

  // GCN aggregation phase (after xw GEMM): selfloop init + edge scatter + bias/act
  auto aggregate = [&](int dout, const int* esrc, const int* edst, const float* dinv,
                       float* outbuf, const float* bb, int relu) {
    selfloop_kernel<<<cdiv_i((long)N * dout, TPB), TPB, 0, stream>>>(bufA, dinv, outbuf, N, dout);
    edge_scatter_kernel<<<4096, TPB, 0, stream>>>(bufA, esrc, edst, dinv, outbuf, E, dout);
    bias_act_kernel<<<cdiv_i((long)N * dout, TPB), TPB, 0, stream>>>(outbuf, bb, N, dout, relu);
  };

  // --- feat branch ---
  run_gemm<31, 32>(feat, W_fc, b_fc, bufF, N, 1, stream);
  run_gemm<32, 32>(bufF, Wf1, nullptr, bufA, N, 0, stream);
  aggregate(32, f_src, f_dst, dinv_f, bufF, bf1, 1);
  run_gemm<32, 33>(bufF, Wf2, nullptr, bufA, N, 0, stream);
  aggregate(33, f_src, f_dst, dinv_f, bufF, bf2, 0);

  // --- spat branch ---
  run_gemm<128, 32>(spat, W_cnn, b_cnn, bufS, N, 1, stream);
  run_gemm<32, 32>(bufS, Ws1, nullptr, bufA, N, 0, stream);
  aggregate(32, s_src, s_dst, dinv_s, bufS, bs1, 1);
  run_gemm<32, 11>(bufS, Ws2, nullptr, bufA, N, 0, stream);
  aggregate(11, s_src, s_dst, dinv_s, bufS, bs2, 0);

  // --- head ---
  concat_relu_kernel<<<cdiv_i((long)N * 44, TPB), TPB, 0, stream>>>(bufF, 33, bufS, 11, bufC, N);
  run_gemm<44, 33>(bufC, Wp1, bp1, bufA, N, 1, stream);
  run_gemm<33, 30>(bufA, Wp2, bp2, (float*)d_out, N, 0, stream);
}